// PGNNLayer_36369783062747
// MI455X (gfx1250) — compile-verified
//
#include <hip/hip_runtime.h>
#include <hip/hip_bf16.h>

#define B_    32
#define N_    4096
#define M_    32
#define INF_  256
#define OUTF_ 256
#define A_    8
#define NR_   (B_ * A_)   // 256 anchor rows

typedef __attribute__((ext_vector_type(2))) float v2f;
typedef __attribute__((ext_vector_type(8))) float v8f;

// ---------------------------------------------------------------------------
// K1: anchor_feat = gather(fatoms at anchor rows) @ W    (256x256 @ 256x256)
// One wave per 16x16 output tile, K-loop of V_WMMA_F32_16X16X4_F32.
// Fragment layouts per CDNA5 ISA 7.12.2:
//   A (16x4 f32):  lanes 0-15 hold M=lane,   VGPR{0,1}=K{0,1};
//                  lanes 16-31 hold M=lane-16, VGPR{0,1}=K{2,3}
//   B (4x16 f32):  lanes 0-15: VGPR{0,1}=rows K{0,1}, col=lane;
//                  lanes 16-31: rows K{2,3}, col=lane-16
//   D (16x16 f32): lanes 0-15: VGPRv = row m0+v;  lanes 16-31: row m0+8+v
// ---------------------------------------------------------------------------
__global__ __launch_bounds__(256) void k_anchor_gemm(
    const float* __restrict__ fatoms, const int* __restrict__ anchor_idx,
    const float* __restrict__ W, float* __restrict__ af)
{
  const int lane = threadIdx.x & 31;
  const int wave = threadIdx.x >> 5;
  const int tile = blockIdx.x * 8 + wave;   // 0..255 (16x16 grid of tiles)
  const int m0 = (tile >> 4) * 16;
  const int n0 = (tile & 15) * 16;
  const int half = lane >> 4;
  const int lm   = lane & 15;

  // Per-lane gathered A-row pointer (anchor row of fatoms)
  const int r   = m0 + lm;                  // output row 0..255
  const int b   = r >> 3;
  const int k   = r & 7;
  const int src = anchor_idx[b * A_ + k];   // 0..N-1
  const float* __restrict__ arow = fatoms + ((size_t)b * N_ + (size_t)src) * INF_;
  const int ncol = n0 + lm;

  v8f acc = {};
  for (int k0 = 0; k0 < INF_; k0 += 4) {
    const int ka = k0 + half * 2;
    v2f av = { arow[ka], arow[ka + 1] };
    v2f bv = { W[ka * OUTF_ + ncol], W[(ka + 1) * OUTF_ + ncol] };
    // 8 args: (neg_a, A, neg_b, B, c_mod, C, reuse_a, reuse_b)
    acc = __builtin_amdgcn_wmma_f32_16x16x4_f32(false, av, false, bv,
                                                (short)0, acc, false, false);
  }

  const int rbase = m0 + half * 8;
#pragma unroll
  for (int v = 0; v < 8; ++v)
    af[(size_t)(rbase + v) * OUTF_ + ncol] = acc[v];
}

// ---------------------------------------------------------------------------
// K2: tiny precompute:
//   block 0: w1 = W @ (a1 + a2)                        (256 floats)
//   block 1: anchor_dot[t] = anchor_feat[t,:] . (a1-a2) (256 floats)
// ---------------------------------------------------------------------------
__global__ __launch_bounds__(256) void k_vectors(
    const float* __restrict__ W, const float* __restrict__ a,
    const float* __restrict__ af, float* __restrict__ w1,
    float* __restrict__ ad)
{
  const int t = threadIdx.x;
  if (blockIdx.x == 0) {
    float s = 0.f;
    for (int j = 0; j < OUTF_; ++j)
      s += W[(size_t)t * OUTF_ + j] * (a[j] + a[OUTF_ + j]);
    w1[t] = s;
  } else {
    float s = 0.f;
    for (int j = 0; j < OUTF_; ++j)
      s += af[(size_t)t * OUTF_ + j] * (a[j] - a[OUTF_ + j]);
    ad[t] = s;
  }
}

// ---------------------------------------------------------------------------
// K3: streaming fused attention. One wave32 per row n:
//   score = fatoms[b,n,:].w1 (coalesced 128B loads + shfl-xor reduce)
//   mask  = ballot(agraph[b,n,lane] == anchor_id[k]) per k
//   softmax over 8, out = sum_k attn_k * anchor_feat[b,k,:]
// ---------------------------------------------------------------------------
__global__ __launch_bounds__(256) void k_attn(
    const float* __restrict__ fatoms, const int* __restrict__ agraph,
    const int* __restrict__ anchor_idx, const float* __restrict__ af,
    const float* __restrict__ w1, const float* __restrict__ ad,
    float* __restrict__ out)
{
  __shared__ float af_s[A_ * OUTF_];
  __shared__ float w1_s[OUTF_];
  __shared__ float ad_s[A_];
  __shared__ int   ai_s[A_];

  const int b = blockIdx.y;
  const int t = threadIdx.x;
  for (int i = t; i < A_ * OUTF_; i += 256)
    af_s[i] = af[(size_t)b * A_ * OUTF_ + i];
  w1_s[t] = w1[t];
  if (t < A_) {
    ad_s[t] = ad[b * A_ + t];
    ai_s[t] = anchor_idx[b * A_ + t];
  }
  __syncthreads();

  const int lane  = t & 31;
  const int wave  = t >> 5;
  const int nbase = blockIdx.x * 32;

  for (int i = 0; i < 4; ++i) {
    const int n = nbase + i * 8 + wave;
    const float* __restrict__ row = fatoms + ((size_t)b * N_ + n) * INF_;
    if (i < 3) __builtin_prefetch(row + 8 * INF_ + lane * 8, 0, 1);

    // score = row . w1  (fully coalesced: lane-stride-1 per load)
    float acc = 0.f;
#pragma unroll
    for (int j = 0; j < 8; ++j)
      acc += row[j * 32 + lane] * w1_s[j * 32 + lane];
#pragma unroll
    for (int off = 16; off; off >>= 1)
      acc += __shfl_xor(acc, off, 32);

    // mask: lane holds one neighbor id; ballot against each anchor
    const int nb = agraph[((size_t)b * N_ + n) * M_ + lane];
    unsigned m8 = 0;
#pragma unroll
    for (int kk = 0; kk < A_; ++kk)
      if (__ballot(nb == ai_s[kk]) != 0ull) m8 |= 1u << kk;

    // softmax over 8 anchors (redundant in every lane; all lanes need attn)
    float e[A_];
    float mx = -3.0e38f;
#pragma unroll
    for (int kk = 0; kk < A_; ++kk) {
      e[kk] = ((m8 >> kk) & 1u) ? (acc + ad_s[kk]) : -1.0e9f;
      mx = fmaxf(mx, e[kk]);
    }
    float ssum = 0.f;
#pragma unroll
    for (int kk = 0; kk < A_; ++kk) { e[kk] = __expf(e[kk] - mx); ssum += e[kk]; }
    const float inv = 1.f / ssum;

    // out[b,n,:] = attn @ anchor_feat   (LDS broadcast, conflict-free)
    float* __restrict__ orow = out + ((size_t)b * N_ + n) * OUTF_;
#pragma unroll
    for (int j = 0; j < 8; ++j) {
      const int col = j * 32 + lane;
      float o = 0.f;
#pragma unroll
      for (int kk = 0; kk < A_; ++kk)
        o += e[kk] * af_s[kk * OUTF_ + col];
      orow[col] = o * inv;
    }
  }
}

// ---------------------------------------------------------------------------
extern "C" void kernel_launch(void* const* d_in, const int* in_sizes, int n_in,
                              void* d_out, int out_size, void* d_ws, size_t ws_size,
                              hipStream_t stream) {
  const float* fatoms     = (const float*)d_in[0];   // (B,N,IN_F) f32
  const int*   agraph     = (const int*)  d_in[1];   // (B,N,M)
  const int*   anchor_idx = (const int*)  d_in[2];   // (B,A)
  const float* W          = (const float*)d_in[3];   // (IN_F,OUT_F)
  const float* a          = (const float*)d_in[4];   // (2*OUT_F,1)
  float* out = (float*)d_out;

  float* af = (float*)d_ws;            // 256*256 floats: anchor_feat
  float* w1 = af + (size_t)NR_ * OUTF_;// 256 floats
  float* ad = w1 + OUTF_;              // 256 floats

  hipLaunchKernelGGL(k_anchor_gemm, dim3(32), dim3(256), 0, stream,
                     fatoms, anchor_idx, W, af);
  hipLaunchKernelGGL(k_vectors, dim3(2), dim3(256), 0, stream,
                     W, a, af, w1, ad);
  hipLaunchKernelGGL(k_attn, dim3(N_ / 32, B_), dim3(256), 0, stream,
                     fatoms, agraph, anchor_idx, af, w1, ad, out);
}